// Electrostatics_56684978372796
// MI455X (gfx1250) — compile-verified
//
#include <hip/hip_runtime.h>

#define N_ATOMS   6144
#define FEAT_DIM  128
#define NT        384                 // 6144 / 16 tiles per dimension
#define TJ_PER_BLK 8                  // 8 waves (wave32) per 256-thread block
#define NTB       (NT / TJ_PER_BLK)   // 48 tile-strips per dimension
#define NPAIR_TB  ((NTB * (NTB + 1)) / 2)       // 1176 (tb, bxj) strip pairs
#define NBLK_PAIR (NPAIR_TB * TJ_PER_BLK)       // 9408 blocks (packed triangle)
#define KE_KCAL   332.0637f
#define R_ON      2.5f                // R_CUT/4
#define INV_DEN   0.2f                // 1/(R_OFF - R_ON) = 1/5

typedef __attribute__((ext_vector_type(2))) float v2f;
typedef __attribute__((ext_vector_type(8))) float v8f;

__device__ __forceinline__ float wave_reduce_sum(float v) {
#pragma unroll
    for (int m = 16; m >= 1; m >>= 1)
        v += __shfl_xor(v, m, 32);
    return v;
}

// ---------------------------------------------------------------------------
// Kernel 1: pred[a] = dot(f[a,:], w) + z_table[z[a]]   (one wave32 per atom)
// ---------------------------------------------------------------------------
__global__ __launch_bounds__(256) void k_pred(const float* __restrict__ f,
                                              const int*   __restrict__ z,
                                              const float* __restrict__ w,
                                              const float* __restrict__ ztab,
                                              float* __restrict__ pred) {
    const int wave = threadIdx.x >> 5;
    const int lane = threadIdx.x & 31;
    const int a    = blockIdx.x * 8 + wave;            // 768 blocks * 8 waves
    const float4 fv = ((const float4*)(f + a * FEAT_DIM))[lane];
    const float4 wv = ((const float4*)w)[lane];
    float dot = fv.x * wv.x + fv.y * wv.y + fv.z * wv.z + fv.w * wv.w;
    dot = wave_reduce_sum(dot);
    if (lane == 0) pred[a] = dot + ztab[z[a]];
}

// ---------------------------------------------------------------------------
// Kernel 2: corr = (total_charge - sum(pred)) / N     (deterministic tree)
// ---------------------------------------------------------------------------
__global__ __launch_bounds__(256) void k_corr(const float* __restrict__ pred,
                                              const float* __restrict__ total_charge,
                                              float* __restrict__ corr) {
    __shared__ float s[256];
    float sum = 0.0f;
    for (int i = threadIdx.x; i < N_ATOMS; i += 256) sum += pred[i];
    s[threadIdx.x] = sum;
    __syncthreads();
    for (int off = 128; off; off >>= 1) {
        if (threadIdx.x < off) s[threadIdx.x] += s[threadIdx.x + off];
        __syncthreads();
    }
    if (threadIdx.x == 0) corr[0] = (total_charge[0] - s[0]) * (1.0f / (float)N_ATOMS);
}

// ---------------------------------------------------------------------------
// Kernel 3: q[i] = pred[i] + corr   (written directly to d_out[1..N])
// ---------------------------------------------------------------------------
__global__ __launch_bounds__(256) void k_q(const float* __restrict__ pred,
                                           const float* __restrict__ corr,
                                           float* __restrict__ qout) {
    const int i = blockIdx.x * 256 + threadIdx.x;
    if (i < N_ATOMS) qout[i] = pred[i] + corr[0];
}

// ---------------------------------------------------------------------------
// Kernel 4: pairwise energy over the packed upper triangle of 16x16 tiles.
//   One tile per wave32. r^2 tile via a single V_WMMA_F32_16X16X4_F32:
//     A row i = (|xi|^2, -2xi, -2yi, -2zi),  B col j = (1, xj, yj, zj),
//     C[m][n] = |xj_n|^2   =>   D = r^2 exactly.
// ---------------------------------------------------------------------------
__device__ __forceinline__ int tri_off(int t) {      // # strip pairs before row t
    return t * NTB - (t * (t - 1)) / 2;
}

__global__ __launch_bounds__(256) void k_pair(const float* __restrict__ xyz,
                                              const float* __restrict__ q,
                                              float* __restrict__ partials) {
    __shared__ float swave[8];
    const int wave = threadIdx.x >> 5;
    const int lane = threadIdx.x & 31;
    const int half = lane >> 4;      // 0: lanes 0-15, 1: lanes 16-31
    const int l15  = lane & 15;

    // ---- packed triangular decode (branchless: estimate + single-step fixup).
    // At strip boundaries 48.5^2 - 2*tri_off(t) == (t-48.5)^2 exactly, so the
    // f32 estimate is within ~5e-6 of an integer; +/-1 corrections suffice.
    const int b   = blockIdx.x;
    const int p   = b >> 3;                       // strip-pair index in [0, 1176)
    const int sub = b & 7;
    int tb = (int)(48.5f - __builtin_amdgcn_sqrtf(48.5f * 48.5f - 2.0f * (float)p));
    tb = tb < 0 ? 0 : (tb > NTB - 1 ? NTB - 1 : tb);
    tb += (tri_off(tb + 1) <= p) ? 1 : 0;
    tb -= (tri_off(tb) > p) ? 1 : 0;
    const int bxj = tb + (p - tri_off(tb));       // column strip, bxj >= tb
    const int ti  = tb * 8 + sub;                 // row tile
    const int tj  = bxj * 8 + wave;               // col tile

    float acc = 0.0f;
    if (tj >= ti) {                               // boundary strips: skip lower waves
        const int i0 = ti * 16, j0 = tj * 16;
        const int irow = i0 + l15, jcol = j0 + l15;

        // ---- batch ALL vmem up front: xyz + q, overlap with packing & WMMA ----
        const float xi = xyz[3 * irow + 0], yi = xyz[3 * irow + 1], zi = xyz[3 * irow + 2];
        const float xj = xyz[3 * jcol + 0], yj = xyz[3 * jcol + 1], zj = xyz[3 * jcol + 2];
        const float qj = q[jcol];
        const int   ibase = i0 + half * 8;
        float qi[8];
#pragma unroll
        for (int r = 0; r < 8; ++r) qi[r] = q[ibase + r];

        const float sqi = xi * xi + yi * yi + zi * zi;
        const float sqj = xj * xj + yj * yj + zj * zj;

        // ISA 32-bit 16x4 A layout: v0 = {K0 | K2}, v1 = {K1 | K3}
        v2f A, B;
        A.x = half ? (-2.0f * yi) : sqi;            // K2 : K0
        A.y = half ? (-2.0f * zi) : (-2.0f * xi);   // K3 : K1
        B.x = half ? yj : 1.0f;                     // K2 : K0 (row-striped over lanes)
        B.y = half ? zj : xj;                       // K3 : K1
        v8f C;
#pragma unroll
        for (int r = 0; r < 8; ++r) C[r] = sqj;     // column-broadcast |xj|^2

        // D[m][n] = r^2 for pair (i0 + r + 8*half, j0 + l15)
        v8f D = __builtin_amdgcn_wmma_f32_16x16x4_f32(
            /*neg_a=*/false, A, /*neg_b=*/false, B,
            /*c_mod=*/(short)0, C, /*reuse_a=*/false, /*reuse_b=*/false);

#pragma unroll
        for (int r = 0; r < 8; ++r) {
            const float d2 = D[r];
            // max(d2,0) > 0  <=>  d2 > 0, and r2_safe only consumed under mask
            const bool  valid = (jcol > (ibase + r)) && (d2 > 0.0f);
            // select the charge product FIRST so the transcendental chain below
            // is unconditional -> compiler keeps it branchless (no saveexec).
            const float qq  = valid ? (qi[r] * qj) : 0.0f;
            const float r2s = valid ? d2 : 1.0f;
            const float rr  = __builtin_amdgcn_sqrtf(r2s);
            const float arg = (rr - R_ON) * INV_DEN;
            const float xu  = 1.0f - arg;
            // sigma(x) = x>0 ? exp(-1/x) : 0
            const float su = (xu  > 0.0f) ? __expf(-__builtin_amdgcn_rcpf(xu))  : 0.0f;
            const float sd = (arg > 0.0f) ? __expf(-__builtin_amdgcn_rcpf(arg)) : 0.0f;
            // fs/sqrt(r2+1) + (1-fs)/r  ==  (su*rsq(r2+1) + sd*rcp(r)) / (su+sd)
            const float num = su * __builtin_amdgcn_rsqf(r2s + 1.0f)
                            + sd * __builtin_amdgcn_rcpf(rr);
            const float term = num * __builtin_amdgcn_rcpf(su + sd);
            acc = fmaf(qq, term, acc);
        }
    }

    acc = wave_reduce_sum(acc);
    if (lane == 0) swave[wave] = acc;
    __syncthreads();
    if (threadIdx.x == 0) {
        float s = 0.0f;
#pragma unroll
        for (int w = 0; w < 8; ++w) s += swave[w];
        partials[b] = s;
    }
}

// ---------------------------------------------------------------------------
// Kernel 5: energy = KE * sum(partials)   (deterministic tree)
// ---------------------------------------------------------------------------
__global__ __launch_bounds__(256) void k_energy(const float* __restrict__ partials,
                                                float* __restrict__ out) {
    __shared__ float s[256];
    float sum = 0.0f;
    for (int i = threadIdx.x; i < NBLK_PAIR; i += 256) sum += partials[i];
    s[threadIdx.x] = sum;
    __syncthreads();
    for (int off = 128; off; off >>= 1) {
        if (threadIdx.x < off) s[threadIdx.x] += s[threadIdx.x + off];
        __syncthreads();
    }
    if (threadIdx.x == 0) out[0] = KE_KCAL * s[0];
}

// ---------------------------------------------------------------------------
extern "C" void kernel_launch(void* const* d_in, const int* in_sizes, int n_in,
                              void* d_out, int out_size, void* d_ws, size_t ws_size,
                              hipStream_t stream) {
    const float* f            = (const float*)d_in[0];
    const int*   z            = (const int*)  d_in[1];
    const float* xyz          = (const float*)d_in[2];
    const float* total_charge = (const float*)d_in[3];
    const float* w_weight     = (const float*)d_in[4];
    const float* z_table      = (const float*)d_in[5];

    float* out      = (float*)d_out;       // [0] = energy, [1..N] = q
    float* ws       = (float*)d_ws;
    float* pred     = ws;                  // N_ATOMS floats
    float* corr     = ws + N_ATOMS;        // 1 float (padded to 16)
    float* partials = ws + N_ATOMS + 16;   // NBLK_PAIR floats
    float* qout     = out + 1;

    k_pred  <<<N_ATOMS / 8, 256, 0, stream>>>(f, z, w_weight, z_table, pred);
    k_corr  <<<1,           256, 0, stream>>>(pred, total_charge, corr);
    k_q     <<<(N_ATOMS + 255) / 256, 256, 0, stream>>>(pred, corr, qout);
    k_pair  <<<NBLK_PAIR,   256, 0, stream>>>(xyz, qout, partials);
    k_energy<<<1,           256, 0, stream>>>(partials, out);
}